// BERTForWordClassification_88433376624823
// MI455X (gfx1250) — compile-verified
//
#include <hip/hip_runtime.h>

// Segment-mean (word pooling) for BERT subword -> word embeddings, on gfx1250.
// sums[w, h] = sum_s 1[word_ids[s] == w] * hidden[s, h]   via V_WMMA_F32_16X16X4_F32
// Sorted word_ids => each 16-word tile's tokens are one contiguous span found by
// binary search, so hidden_states streams from HBM exactly once (~300 MB total,
// ~13 us roofline at 23.3 TB/s). Two WMMA accumulators hide XDL latency.

typedef float v2f __attribute__((ext_vector_type(2)));
typedef float v8f __attribute__((ext_vector_type(8)));

#define BATCH   16
#define SEQ     4096
#define HID     768
#define NWORDS  2048

__global__ __launch_bounds__(256)
void segmean_wmma_kernel(const float* __restrict__ hidden,
                         const int*   __restrict__ wids,
                         float*       __restrict__ out)
{
    __shared__ int   lb[17];     // lower bounds for words w0 .. w0+16
    __shared__ float rcnt[16];   // 1 / max(count, 1) per word in tile

    const int b     = blockIdx.z;            // sample
    const int w0    = blockIdx.y << 4;       // first word of this 16-word tile
    const int hbase = blockIdx.x << 7;       // first hidden dim of this 128-wide slab
    const int tid   = threadIdx.x;

    const int* __restrict__ wrow = wids + b * SEQ;

    // ---- per-block: 17 binary searches over sorted word_ids[1 .. SEQ-2] ----
    if (tid < 17) {
        const int v  = w0 + tid;
        int lo = 1, hi = SEQ - 1;            // wids[0] == wids[SEQ-1] == -1 (specials)
        while (lo < hi) {
            const int mid = (lo + hi) >> 1;
            if (wrow[mid] < v) lo = mid + 1; else hi = mid;
        }
        lb[tid] = lo;
    }
    __syncthreads();

    if (tid < 16) {
        float cnt = (float)(lb[tid + 1] - lb[tid]);
        if (cnt < 1.0f) cnt = 1.0f;          // empty words stay 0 (0 * 1)
        rcnt[tid] = __builtin_amdgcn_rcpf(cnt);   // single v_rcp_f32
    }

    const int start = lb[0];
    const int end   = lb[16];

    // ---- wave / lane decomposition (wave32) ----
    const int wave  = tid >> 5;              // 8 waves -> 8 h-tiles of 16
    const int laneN = tid & 15;              // N index (hidden col within tile)
    const int laneH = (tid >> 4) & 1;        // 0: lanes 0-15, 1: lanes 16-31
    const int h0    = hbase + (wave << 4);

    const int myw = w0 + laneN;              // A-matrix row this lane supplies
    const float* __restrict__ hb = hidden + (size_t)b * SEQ * HID + h0 + laneN;

    v8f c0 = {0.f, 0.f, 0.f, 0.f, 0.f, 0.f, 0.f, 0.f};
    v8f c1 = c0;

    __syncthreads();                         // rcnt visible before any early-exit paths

    // ---- main loop: 8 tokens / 2 independent WMMAs per iteration ----
    // A 16x4 f32 layout: lanes 0-15 hold K=0(v0),K=1(v1); lanes 16-31 hold K=2,K=3.
    // B 4x16 mirrors the K split across lane halves. Clamped rows have wid=-1 -> 0.
    for (int s0 = (start & ~7); s0 < end; s0 += 8) {
        int sa = s0 + (laneH << 1);
        int sb = sa + 1;
        int sc = sa + 4;
        int sd = sb + 4;
        if (sa > SEQ - 1) sa = SEQ - 1;
        if (sb > SEQ - 1) sb = SEQ - 1;
        if (sc > SEQ - 1) sc = SEQ - 1;
        if (sd > SEQ - 1) sd = SEQ - 1;

        const int wa = wrow[sa];
        const int wb = wrow[sb];
        const int wc = wrow[sc];
        const int wd = wrow[sd];

        v2f a0, a1, b0, b1;
        a0.x = (wa == myw) ? 1.0f : 0.0f;
        a0.y = (wb == myw) ? 1.0f : 0.0f;
        a1.x = (wc == myw) ? 1.0f : 0.0f;
        a1.y = (wd == myw) ? 1.0f : 0.0f;
        b0.x = __builtin_nontemporal_load(hb + (size_t)sa * HID);
        b0.y = __builtin_nontemporal_load(hb + (size_t)sb * HID);
        b1.x = __builtin_nontemporal_load(hb + (size_t)sc * HID);
        b1.y = __builtin_nontemporal_load(hb + (size_t)sd * HID);

        c0 = __builtin_amdgcn_wmma_f32_16x16x4_f32(
                false, a0, false, b0, (short)0, c0, false, false);
        c1 = __builtin_amdgcn_wmma_f32_16x16x4_f32(
                false, a1, false, b1, (short)0, c1, false, false);
    }

    const v8f c = c0 + c1;

    // ---- epilogue: mean via precomputed reciprocal, streamed 16x16 f32 store ----
    // C/D layout: VGPR r -> row (r + 8*laneH), col laneN.
    float* __restrict__ ob = out + ((size_t)b * NWORDS + w0) * HID + h0 + laneN;
    #pragma unroll
    for (int r = 0; r < 8; ++r) {
        const int widx = r + (laneH << 3);
        __builtin_nontemporal_store(c[r] * rcnt[widx], ob + (size_t)widx * HID);
    }
}

extern "C" void kernel_launch(void* const* d_in, const int* in_sizes, int n_in,
                              void* d_out, int out_size, void* d_ws, size_t ws_size,
                              hipStream_t stream) {
    const float* hidden   = (const float*)d_in[0];
    const int*   word_ids = (const int*)  d_in[1];
    float*       out      = (float*)      d_out;

    dim3 grid(HID / 128, NWORDS / 16, BATCH);   // (6, 128, 16)
    segmean_wmma_kernel<<<grid, 256, 0, stream>>>(hidden, word_ids, out);
}